// RecurrentGCN1_37769942401404
// MI455X (gfx1250) — compile-verified
//
#include <hip/hip_runtime.h>
#include <hip/hip_bf16.h>

typedef __attribute__((ext_vector_type(2))) float v2f;
typedef __attribute__((ext_vector_type(8))) float v8f;

#define FDIM 128
#define HCH  4

// ---------------------------------------------------------------------------
// helpers
// ---------------------------------------------------------------------------
__device__ __forceinline__ float sigmoidf_(float x) {
    return 1.0f / (1.0f + __expf(-x));
}

__device__ __forceinline__ unsigned f2key(float f) {
    unsigned u = __float_as_uint(f);
    return (u >> 31) ? ~u : (u | 0x80000000u);
}
__device__ __forceinline__ float key2f(unsigned u) {
    return (u & 0x80000000u) ? __uint_as_float(u ^ 0x80000000u)
                             : __uint_as_float(~u);
}

__device__ __forceinline__ v8f wmma4(v2f a, v2f b, v8f c) {
    // V_WMMA_F32_16X16X4_F32: D = A(16x4) x B(4x16) + C(16x16)
    return __builtin_amdgcn_wmma_f32_16x16x4_f32(
        false, a, false, b, (short)0, c, false, false);
}

// async global->LDS copy (CDNA5 GLOBAL_LOAD_ASYNC_TO_LDS_B128, ASYNCcnt)
#if __has_builtin(__builtin_amdgcn_global_load_async_to_lds_b128)
#define HAVE_ASYNC_LDS 1
typedef int v4i_ __attribute__((__vector_size__(16)));
typedef __attribute__((address_space(1))) v4i_ g_v4i;   // global (AS1) v4i
typedef __attribute__((address_space(3))) v4i_ l_v4i;   // LDS (AS3) v4i
#else
#define HAVE_ASYNC_LDS 0
#endif

__device__ __forceinline__ void wait_async_zero() {
#if __has_builtin(__builtin_amdgcn_s_wait_asynccnt)
    __builtin_amdgcn_s_wait_asynccnt(0);
#else
    asm volatile("s_wait_asynccnt 0" ::: "memory");
#endif
}

// state layout (u32): [0]=prefix [1]=krem [2]=cntHigh [3]=cntEq [8..263]=hist
#define ST_PREFIX 0
#define ST_KREM   1
#define ST_CHI    2
#define ST_CEQ    3
#define ST_HIST   8

// ---------------------------------------------------------------------------
// 0. ||p||^{-1}
// ---------------------------------------------------------------------------
__global__ void pnorm_kernel(const float* __restrict__ p, float* __restrict__ pinv) {
    __shared__ float red[128];
    float v = p[threadIdx.x];
    red[threadIdx.x] = v * v;
    __syncthreads();
    for (int s = 64; s > 0; s >>= 1) {
        if (threadIdx.x < s) red[threadIdx.x] += red[threadIdx.x + s];
        __syncthreads();
    }
    if (threadIdx.x == 0) pinv[0] = rsqrtf(red[0]);
}

// ---------------------------------------------------------------------------
// 1. score = x @ p / ||p||  -> sortable key per node (wave per node)
// ---------------------------------------------------------------------------
__global__ void score_kernel(const float* __restrict__ x, const float* __restrict__ p,
                             const float* __restrict__ pinv,
                             unsigned* __restrict__ keys, int n) {
    int wave = threadIdx.x >> 5, lane = threadIdx.x & 31;
    int i = blockIdx.x * 8 + wave;
    if (i >= n) return;
    float s = 0.f;
    #pragma unroll
    for (int k = lane; k < FDIM; k += 32) s += x[(size_t)i * FDIM + k] * p[k];
    #pragma unroll
    for (int off = 16; off > 0; off >>= 1) s += __shfl_xor(s, off, 32);
    if (lane == 0) keys[i] = f2key(s * pinv[0]);
}

// ---------------------------------------------------------------------------
// 2. radix-select top-128 (4 passes over MSB..LSB bytes)
// ---------------------------------------------------------------------------
__global__ void select_init_kernel(unsigned* st) {
    for (int t = threadIdx.x; t < ST_HIST + 256; t += 256) st[t] = 0;
    if (threadIdx.x == 0) st[ST_KREM] = FDIM;
}

__global__ void hist_kernel(const unsigned* __restrict__ keys, int n,
                            unsigned* __restrict__ st, int pass) {
    int i = blockIdx.x * 256 + threadIdx.x;
    if (i >= n) return;
    unsigned u = keys[i];
    int shift = 24 - 8 * pass;
    if (pass == 0 || ((u ^ st[ST_PREFIX]) >> (shift + 8)) == 0)
        atomicAdd(&st[ST_HIST + ((u >> shift) & 0xFF)], 1u);
}

__global__ void scan_kernel(unsigned* st, int pass) {
    int shift = 24 - 8 * pass;
    unsigned k = st[ST_KREM];
    unsigned cum = 0;
    int chosen = 0;
    for (int b = 255; b >= 0; --b) {
        unsigned c = st[ST_HIST + b];
        if (cum + c >= k) { chosen = b; break; }
        cum += c;
    }
    st[ST_PREFIX] |= ((unsigned)chosen) << shift;
    st[ST_KREM] = k - cum;
    for (int b = 0; b < 256; ++b) st[ST_HIST + b] = 0;
}

__global__ void compact_kernel(const unsigned* __restrict__ keys, int n,
                               unsigned* __restrict__ st,
                               unsigned* __restrict__ hiKey, unsigned* __restrict__ hiIdx,
                               unsigned* __restrict__ eqIdx) {
    int i = blockIdx.x * 256 + threadIdx.x;
    if (i >= n) return;
    unsigned u = keys[i], T = st[ST_PREFIX];
    if (u > T) {
        unsigned pos = atomicAdd(&st[ST_CHI], 1u);
        if (pos < FDIM) { hiKey[pos] = u; hiIdx[pos] = (unsigned)i; }
    } else if (u == T) {
        unsigned pos = atomicAdd(&st[ST_CEQ], 1u);
        if (pos < FDIM) eqIdx[pos] = (unsigned)i;
    }
}

__global__ void topk_finalize_kernel(const unsigned* __restrict__ st,
                                     const unsigned* __restrict__ hiKey,
                                     const unsigned* __restrict__ hiIdx,
                                     const unsigned* __restrict__ eqIdx,
                                     int* __restrict__ perm, float* __restrict__ vals) {
    __shared__ unsigned sk[FDIM];
    __shared__ unsigned si[FDIM];
    int t = threadIdx.x;
    unsigned T   = st[ST_PREFIX];
    unsigned nHi = st[ST_CHI]; if (nHi > FDIM) nHi = FDIM;
    unsigned nEq = st[ST_CEQ]; if (nEq > FDIM) nEq = FDIM;
    unsigned k, id;
    if ((unsigned)t < nHi) { k = hiKey[t]; id = hiIdx[t]; }
    else {
        unsigned e = (unsigned)t - nHi;
        id = (e < nEq) ? eqIdx[e] : 0u;
        k  = T;
    }
    sk[t] = k; si[t] = id;
    // bitonic sort, descending by key
    for (int len = 2; len <= FDIM; len <<= 1) {
        for (int strd = len >> 1; strd > 0; strd >>= 1) {
            __syncthreads();
            int j = t ^ strd;
            if (j > t) {
                bool desc = ((t & len) == 0);
                unsigned a = sk[t], b = sk[j];
                bool sw = desc ? (a < b) : (a > b);
                if (sw) {
                    sk[t] = b; sk[j] = a;
                    unsigned tmp = si[t]; si[t] = si[j]; si[j] = tmp;
                }
            }
        }
    }
    __syncthreads();
    perm[t] = (int)si[t];
    vals[t] = key2f(sk[t]);
}

// ---------------------------------------------------------------------------
// 3. x_tilde[r][c] = x[perm[r]][c] * tanh(vals[r])
// ---------------------------------------------------------------------------
__global__ void xtilde_kernel(const float* __restrict__ x,
                              const int* __restrict__ perm,
                              const float* __restrict__ vals,
                              float* __restrict__ xt) {
    int r = blockIdx.x, c = threadIdx.x;
    xt[r * FDIM + c] = x[(size_t)perm[r] * FDIM + c] * tanhf(vals[r]);
}

// ---------------------------------------------------------------------------
// 4. GRU: W_new = GRUCell(x_tilde, W_evolve)  -- 6 fused WMMA GEMMs/tile
//    one 16x16 output tile per wave; 64 tiles total
// ---------------------------------------------------------------------------
__global__ void gru_kernel(const float* __restrict__ xt, const float* __restrict__ Wev,
                           const float* __restrict__ W_ih, const float* __restrict__ W_hh,
                           const float* __restrict__ b_ih, const float* __restrict__ b_hh,
                           float* __restrict__ Wn) {
    int wave = threadIdx.x >> 5, lane = threadIdx.x & 31;
    int tile = blockIdx.x * 8 + wave;
    int m0 = (tile >> 3) * 16;
    int n0 = (tile & 7) * 16;
    int half = lane >> 4, lid = lane & 15;
    int col = n0 + lid;

    v8f ir = {}, iz = {}, in_ = {}, hr = {}, hz = {}, hn = {};
    for (int k0 = 0; k0 < FDIM; k0 += 4) {
        int ka = k0 + 2 * half;
        v2f ax, ah;
        ax.x = xt [(m0 + lid) * FDIM + ka];  ax.y = xt [(m0 + lid) * FDIM + ka + 1];
        ah.x = Wev[(m0 + lid) * FDIM + ka];  ah.y = Wev[(m0 + lid) * FDIM + ka + 1];
        // B = W^T : B[k][n] = W[n*F + k]; gate row offsets 0/128/256
        v2f br, bz, bn, cr, cz, cn;
        br.x = W_ih[(col      ) * FDIM + ka]; br.y = W_ih[(col      ) * FDIM + ka + 1];
        bz.x = W_ih[(col + 128) * FDIM + ka]; bz.y = W_ih[(col + 128) * FDIM + ka + 1];
        bn.x = W_ih[(col + 256) * FDIM + ka]; bn.y = W_ih[(col + 256) * FDIM + ka + 1];
        cr.x = W_hh[(col      ) * FDIM + ka]; cr.y = W_hh[(col      ) * FDIM + ka + 1];
        cz.x = W_hh[(col + 128) * FDIM + ka]; cz.y = W_hh[(col + 128) * FDIM + ka + 1];
        cn.x = W_hh[(col + 256) * FDIM + ka]; cn.y = W_hh[(col + 256) * FDIM + ka + 1];
        ir  = wmma4(ax, br, ir);
        iz  = wmma4(ax, bz, iz);
        in_ = wmma4(ax, bn, in_);
        hr  = wmma4(ah, cr, hr);
        hz  = wmma4(ah, cz, hz);
        hn  = wmma4(ah, cn, hn);
    }
    #pragma unroll
    for (int v = 0; v < 8; ++v) {
        int row = m0 + v + 8 * half;
        float r  = sigmoidf_(ir[v] + b_ih[col      ] + hr[v] + b_hh[col      ]);
        float z  = sigmoidf_(iz[v] + b_ih[col + 128] + hz[v] + b_hh[col + 128]);
        float nn = tanhf(in_[v] + b_ih[col + 256] + r * (hn[v] + b_hh[col + 256]));
        Wn[row * FDIM + col] = (1.0f - z) * nn + z * Wev[row * FDIM + col];
    }
}

// ---------------------------------------------------------------------------
// 5. xw = x @ W_new   (N x 128 x 128), 16x16 tile per wave
//    x tile staged in LDS via GLOBAL_LOAD_ASYNC_TO_LDS_B128 (ASYNCcnt)
// ---------------------------------------------------------------------------
__global__ void xw_kernel(const float* __restrict__ x, const float* __restrict__ Wn,
                          float* __restrict__ xw) {
    __shared__ float xs[16 * FDIM];
    int row0 = blockIdx.x * 16;

#if HAVE_ASYNC_LDS
    // 16*128 floats = 512 x 16B chunks; 256 threads x 2 chunks, no VGPR staging
    #pragma unroll
    for (int i = threadIdx.x; i < 512; i += 256) {
        const float* gp = x + (size_t)row0 * FDIM + (size_t)i * 4;
        float*       lp = &xs[i * 4];
        __builtin_amdgcn_global_load_async_to_lds_b128(
            (g_v4i*)gp, (l_v4i*)lp, 0, 0);
    }
    wait_async_zero();
#else
    for (int i = threadIdx.x; i < 16 * FDIM; i += 256)
        xs[i] = x[(size_t)row0 * FDIM + i];
#endif
    __builtin_prefetch(x + (size_t)(row0 + 16) * FDIM, 0, 0);
    __syncthreads();

    int wave = threadIdx.x >> 5, lane = threadIdx.x & 31;
    int n0 = wave * 16;
    int half = lane >> 4, lid = lane & 15;
    v8f acc = {};
    for (int k0 = 0; k0 < FDIM; k0 += 4) {
        int ka = k0 + 2 * half;
        v2f a, b;
        a.x = xs[lid * FDIM + ka];
        a.y = xs[lid * FDIM + ka + 1];
        b.x = Wn[(ka)     * FDIM + n0 + lid];
        b.y = Wn[(ka + 1) * FDIM + n0 + lid];
        acc = wmma4(a, b, acc);
    }
    #pragma unroll
    for (int v = 0; v < 8; ++v) {
        int row = row0 + v + 8 * half;
        xw[(size_t)row * FDIM + n0 + lid] = acc[v];
    }
}

// ---------------------------------------------------------------------------
// 6. degrees / norms
// ---------------------------------------------------------------------------
__global__ void deg_init_kernel(float* deg0, float* deg1, int n) {
    int i = blockIdx.x * 256 + threadIdx.x;
    if (i < n) { deg0[i] = 1.0f; deg1[i] = 1.0f; }   // self-loop weight
}
__global__ void deg_edge_kernel(const long long* __restrict__ ei,
                                const float* __restrict__ ew,
                                float* deg0, float* deg1, int nE) {
    int e = blockIdx.x * 256 + threadIdx.x;
    if (e >= nE) return;
    int d = (int)ei[(size_t)nE + e];
    atomicAdd(&deg0[d], ew[e]);
    atomicAdd(&deg1[d], 1.0f);
}
__global__ void dinv_kernel(float* deg0, float* deg1, int n) {
    int i = blockIdx.x * 256 + threadIdx.x;
    if (i >= n) return;
    float a = deg0[i]; deg0[i] = (a > 0.f) ? rsqrtf(a) : 0.f;
    float b = deg1[i]; deg1[i] = (b > 0.f) ? rsqrtf(b) : 0.f;
}

// ---------------------------------------------------------------------------
// 7. propagate 1 (128 channels): self-loop init + wave-per-edge scatter
// ---------------------------------------------------------------------------
__global__ void prop1_init_kernel(const float* __restrict__ xw,
                                  const float* __restrict__ dinv0,
                                  float* __restrict__ outh, size_t total) {
    size_t t = (size_t)blockIdx.x * 256 + threadIdx.x;
    if (t >= total) return;
    size_t i = t >> 7;
    float di = dinv0[i];
    outh[t] = xw[t] * di * di;
}
__global__ void prop1_edge_kernel(const long long* __restrict__ ei,
                                  const float* __restrict__ ew,
                                  const float* __restrict__ dinv0,
                                  const float* __restrict__ xw,
                                  float* __restrict__ outh, int nE) {
    int wave = threadIdx.x >> 5, lane = threadIdx.x & 31;
    int e = blockIdx.x * 8 + wave;
    if (e >= nE) return;
    int s = (int)ei[e];
    int d = (int)ei[(size_t)nE + e];
    float nrm = dinv0[s] * ew[e] * dinv0[d];
    #pragma unroll
    for (int c = lane; c < FDIM; c += 32)
        atomicAdd(&outh[(size_t)d * FDIM + c], xw[(size_t)s * FDIM + c] * nrm);
}

// ---------------------------------------------------------------------------
// 8. z = h @ conv1_W   (wave per node, 4 outputs)
// ---------------------------------------------------------------------------
__global__ void conv1_kernel(const float* __restrict__ h, const float* __restrict__ cw,
                             float* __restrict__ z, int n) {
    int wave = threadIdx.x >> 5, lane = threadIdx.x & 31;
    int i = blockIdx.x * 8 + wave;
    if (i >= n) return;
    float acc0 = 0.f, acc1 = 0.f, acc2 = 0.f, acc3 = 0.f;
    for (int k = lane; k < FDIM; k += 32) {
        float hv = h[(size_t)i * FDIM + k];
        acc0 += hv * cw[k * HCH + 0];
        acc1 += hv * cw[k * HCH + 1];
        acc2 += hv * cw[k * HCH + 2];
        acc3 += hv * cw[k * HCH + 3];
    }
    #pragma unroll
    for (int off = 16; off > 0; off >>= 1) {
        acc0 += __shfl_xor(acc0, off, 32);
        acc1 += __shfl_xor(acc1, off, 32);
        acc2 += __shfl_xor(acc2, off, 32);
        acc3 += __shfl_xor(acc3, off, 32);
    }
    if (lane == 0) {
        z[(size_t)i * HCH + 0] = acc0;
        z[(size_t)i * HCH + 1] = acc1;
        z[(size_t)i * HCH + 2] = acc2;
        z[(size_t)i * HCH + 3] = acc3;
    }
}

// ---------------------------------------------------------------------------
// 9. propagate 2 (4 channels, ew = 1)
// ---------------------------------------------------------------------------
__global__ void prop2_init_kernel(const float* __restrict__ z,
                                  const float* __restrict__ dinv1,
                                  float* __restrict__ out2, size_t total) {
    size_t t = (size_t)blockIdx.x * 256 + threadIdx.x;
    if (t >= total) return;
    size_t i = t >> 2;
    float di = dinv1[i];
    out2[t] = z[t] * di * di;
}
__global__ void prop2_edge_kernel(const long long* __restrict__ ei,
                                  const float* __restrict__ dinv1,
                                  const float* __restrict__ z,
                                  float* __restrict__ out2, int nE) {
    int e = blockIdx.x * 256 + threadIdx.x;
    if (e >= nE) return;
    int s = (int)ei[e];
    int d = (int)ei[(size_t)nE + e];
    float nrm = dinv1[s] * dinv1[d];
    #pragma unroll
    for (int c = 0; c < HCH; ++c)
        atomicAdd(&out2[(size_t)d * HCH + c], z[(size_t)s * HCH + c] * nrm);
}

// ---------------------------------------------------------------------------
// 10. epilogue: h = out2 + b; out = tanh(h); h_lin = h @ lin_W.T + lin_b
// ---------------------------------------------------------------------------
__global__ void final_kernel(const float* __restrict__ out2,
                             const float* __restrict__ cb,
                             const float* __restrict__ lw,
                             const float* __restrict__ lb,
                             float* __restrict__ d_out, int n) {
    int i = blockIdx.x * 256 + threadIdx.x;
    if (i >= n) return;
    float hl = lb[0];
    #pragma unroll
    for (int c = 0; c < HCH; ++c) {
        float hc = out2[(size_t)i * HCH + c] + cb[c];
        d_out[(size_t)n + (size_t)i * HCH + c] = tanhf(hc);
        hl += hc * lw[c];
    }
    d_out[i] = hl;   // h_lin [N,1] first in tuple order
}

// ---------------------------------------------------------------------------
// launch
// ---------------------------------------------------------------------------
extern "C" void kernel_launch(void* const* d_in, const int* in_sizes, int n_in,
                              void* d_out, int out_size, void* d_ws, size_t ws_size,
                              hipStream_t stream) {
    const float*     x    = (const float*)d_in[0];
    const long long* ei   = (const long long*)d_in[1];   // int64 edge_index
    const float*     ew   = (const float*)d_in[2];
    const float*     p    = (const float*)d_in[3];
    const float*     W_ih = (const float*)d_in[4];
    const float*     W_hh = (const float*)d_in[5];
    const float*     b_ih = (const float*)d_in[6];
    const float*     b_hh = (const float*)d_in[7];
    const float*     Wev  = (const float*)d_in[8];
    const float*     cw   = (const float*)d_in[9];
    const float*     cb   = (const float*)d_in[10];
    const float*     lw   = (const float*)d_in[11];
    const float*     lb   = (const float*)d_in[12];

    const int N = in_sizes[0] / FDIM;
    const int E = in_sizes[2];

    // workspace carve-out
    char* base = (char*)d_ws;
    size_t off = 0;
    auto carve = [&](size_t bytes) -> char* {
        char* q = base + off;
        off = (off + bytes + 255) & ~(size_t)255;
        return q;
    };
    unsigned* keys  = (unsigned*)carve((size_t)N * 4);
    unsigned* st    = (unsigned*)carve((ST_HIST + 256) * 4);
    unsigned* hiKey = (unsigned*)carve(FDIM * 4);
    unsigned* hiIdx = (unsigned*)carve(FDIM * 4);
    unsigned* eqIdx = (unsigned*)carve(FDIM * 4);
    int*      perm  = (int*)carve(FDIM * 4);
    float*    vals  = (float*)carve(FDIM * 4);
    float*    pinv  = (float*)carve(256);
    float*    xt    = (float*)carve((size_t)FDIM * FDIM * 4);
    float*    Wn    = (float*)carve((size_t)FDIM * FDIM * 4);
    float*    deg0  = (float*)carve((size_t)N * 4);
    float*    deg1  = (float*)carve((size_t)N * 4);
    float*    xw    = (float*)carve((size_t)N * FDIM * 4);
    float*    outh  = (float*)carve((size_t)N * FDIM * 4);
    float*    z     = (float*)carve((size_t)N * HCH * 4);
    float*    out2  = (float*)carve((size_t)N * HCH * 4);
    (void)ws_size; (void)n_in; (void)out_size;

    // --- scoring + top-k ---
    pnorm_kernel<<<1, 128, 0, stream>>>(p, pinv);
    score_kernel<<<(N + 7) / 8, 256, 0, stream>>>(x, p, pinv, keys, N);
    select_init_kernel<<<1, 256, 0, stream>>>(st);
    for (int pass = 0; pass < 4; ++pass) {
        hist_kernel<<<(N + 255) / 256, 256, 0, stream>>>(keys, N, st, pass);
        scan_kernel<<<1, 1, 0, stream>>>(st, pass);
    }
    compact_kernel<<<(N + 255) / 256, 256, 0, stream>>>(keys, N, st, hiKey, hiIdx, eqIdx);
    topk_finalize_kernel<<<1, FDIM, 0, stream>>>(st, hiKey, hiIdx, eqIdx, perm, vals);
    xtilde_kernel<<<FDIM, FDIM, 0, stream>>>(x, perm, vals, xt);

    // --- GRU weight evolution (WMMA) ---
    gru_kernel<<<8, 256, 0, stream>>>(xt, Wev, W_ih, W_hh, b_ih, b_hh, Wn);

    // --- big GEMM xw = x @ W_new (WMMA + async LDS staging) ---
    xw_kernel<<<N / 16, 256, 0, stream>>>(x, Wn, xw);

    // --- degrees / norms ---
    deg_init_kernel<<<(N + 255) / 256, 256, 0, stream>>>(deg0, deg1, N);
    deg_edge_kernel<<<(E + 255) / 256, 256, 0, stream>>>(ei, ew, deg0, deg1, E);
    dinv_kernel<<<(N + 255) / 256, 256, 0, stream>>>(deg0, deg1, N);

    // --- propagate 1 (128 ch) ---
    size_t tot1 = (size_t)N * FDIM;
    prop1_init_kernel<<<(unsigned)((tot1 + 255) / 256), 256, 0, stream>>>(xw, deg0, outh, tot1);
    prop1_edge_kernel<<<(E + 7) / 8, 256, 0, stream>>>(ei, ew, deg0, xw, outh, E);

    // --- conv1 GEMV + propagate 2 (4 ch) ---
    conv1_kernel<<<(N + 7) / 8, 256, 0, stream>>>(outh, cw, z, N);
    size_t tot2 = (size_t)N * HCH;
    prop2_init_kernel<<<(unsigned)((tot2 + 255) / 256), 256, 0, stream>>>(z, deg1, out2, tot2);
    prop2_edge_kernel<<<(E + 255) / 256, 256, 0, stream>>>(ei, deg1, z, out2, E);

    // --- epilogue ---
    final_kernel<<<(N + 255) / 256, 256, 0, stream>>>(out2, cb, lw, lb, (float*)d_out, N);
}